// GNNSentimentClassifier_46634754900429
// MI455X (gfx1250) — compile-verified
//
#include <hip/hip_runtime.h>
#include <hip/hip_bf16.h>
#include <stddef.h>

// ---------------------------------------------------------------------------
// GCN forward for MI455X (gfx1250, wave32, WMMA f32, async LDS staging).
// ---------------------------------------------------------------------------

typedef float v8f __attribute__((ext_vector_type(8)));
typedef float v2f __attribute__((ext_vector_type(2)));

#define HID 128   // hidden width (N dimension of every GEMM)
#define TM  128   // rows per block (8 waves x 16)
#define KC  32    // K chunk staged in LDS (double-buffered)
#define APAD 36   // padded A row stride in floats: 144B -> 16B aligned,
                  // 36*lrow mod 64 spreads 16 lanes over distinct banks

#if defined(__gfx1250__) && __has_builtin(__builtin_amdgcn_global_load_async_to_lds_b128)
#define HAVE_ASYNC_LDS 1
typedef int v4i __attribute__((vector_size(16)));
typedef __attribute__((address_space(1))) v4i* gas_ptr;   // global operand
typedef __attribute__((address_space(3))) v4i* las_ptr;   // LDS operand
__device__ __forceinline__ void async_wait0() {
#if __has_builtin(__builtin_amdgcn_s_wait_asynccnt)
    __builtin_amdgcn_s_wait_asynccnt(0);
#else
    asm volatile("s_wait_asynccnt 0" ::: "memory");
#endif
}
#endif

// ---------------------------------------------------------------------------
// C[M x 128] = A[M x K] @ B[K x 128] (+ bias) (+ relu), WMMA f32 16x16x4.
// 256 threads = 8 wave32; wave w owns rows [m0+16w, +16), all 128 cols.
// Double-buffered K-chunks: async A -> LDS, B global loads in flight during
// compute, reformatted into K-pair-interleaved LDS for single-b64 fragments.
// ---------------------------------------------------------------------------
__global__ __launch_bounds__(256)
void wmma_gemm_f32(const float* __restrict__ A, const float* __restrict__ B,
                   const float* __restrict__ bias, float* __restrict__ C,
                   int M, int K, int use_bias, int use_relu)
{
    __shared__ float As[2][TM * APAD];        // 2 x 18 KB
    __shared__ float Bp[2][(KC / 2) * 256];   // 2 x 16 KB, pair-interleaved

    const int tid   = threadIdx.x;
    const int wave  = tid >> 5;               // 0..7
    const int lane  = tid & 31;
    const int lrow  = lane & 15;              // M (A) / N (B) within tile
    const int khalf = (lane >> 4) * 2;        // lanes 0-15 -> K{0,1}; 16-31 -> K{2,3}
    const int m0    = blockIdx.x * TM;

    v8f acc[8];
    #pragma unroll
    for (int n = 0; n < 8; ++n)
        #pragma unroll
        for (int i = 0; i < 8; ++i) acc[n][i] = 0.0f;

    // A tile: TM x KC = 1024 float4 -> 4 per thread
    // B tile: KC x 128 = 1024 float4 -> 4 per thread
    const int nchunks = K / KC;
    float4 breg[4];

    // ---- stage chunk 0 ----
    #pragma unroll
    for (int i = 0; i < 4; ++i) {
        int f  = tid + i * 256;
        int r  = f >> 3;                       // 8 float4 per A row
        int c4 = f & 7;
        int gr = m0 + r; if (gr >= M) gr = M - 1;
#ifdef HAVE_ASYNC_LDS
        __builtin_amdgcn_global_load_async_to_lds_b128(
            (gas_ptr)(A + (size_t)gr * K + c4 * 4),
            (las_ptr)(&As[0][r * APAD + c4 * 4]), 0, 0);
#else
        *(float4*)(&As[0][r * APAD + c4 * 4]) =
            *(const float4*)(A + (size_t)gr * K + c4 * 4);
#endif
    }
    #pragma unroll
    for (int i = 0; i < 4; ++i) {
        int f  = tid + i * 256;
        int k  = f >> 5;                       // 32 float4 per B row
        int c4 = f & 31;
        breg[i] = *(const float4*)(B + (size_t)k * HID + c4 * 4);
    }
    #pragma unroll
    for (int i = 0; i < 4; ++i) {              // reformat: Bp[k/2][2*col + (k&1)]
        int f  = tid + i * 256;
        int k  = f >> 5;
        int c4 = f & 31;
        float* d = &Bp[0][(k >> 1) * 256 + (k & 1)];
        d[2 * (c4 * 4 + 0)] = breg[i].x;
        d[2 * (c4 * 4 + 1)] = breg[i].y;
        d[2 * (c4 * 4 + 2)] = breg[i].z;
        d[2 * (c4 * 4 + 3)] = breg[i].w;
    }
#ifdef HAVE_ASYNC_LDS
    async_wait0();
#endif
    __syncthreads();

    for (int c = 0; c < nchunks; ++c) {
        const int cur = c & 1, nxt = cur ^ 1;
        const int kc_n = (c + 1) * KC;
        const bool more = (c + 1) < nchunks;

        // ---- kick off next chunk: async A -> LDS buffer nxt, B -> registers ----
        if (more) {
            #pragma unroll
            for (int i = 0; i < 4; ++i) {
                int f  = tid + i * 256;
                int r  = f >> 3;
                int c4 = f & 7;
                int gr = m0 + r; if (gr >= M) gr = M - 1;
#ifdef HAVE_ASYNC_LDS
                __builtin_amdgcn_global_load_async_to_lds_b128(
                    (gas_ptr)(A + (size_t)gr * K + kc_n + c4 * 4),
                    (las_ptr)(&As[nxt][r * APAD + c4 * 4]), 0, 0);
#else
                *(float4*)(&As[nxt][r * APAD + c4 * 4]) =
                    *(const float4*)(A + (size_t)gr * K + kc_n + c4 * 4);
#endif
            }
            #pragma unroll
            for (int i = 0; i < 4; ++i) {
                int f  = tid + i * 256;
                int k  = f >> 5;
                int c4 = f & 31;
                breg[i] = *(const float4*)(B + (size_t)(kc_n + k) * HID + c4 * 4);
            }
        }

        // ---- compute on current buffers: 8 K-steps x 8 N-tiles ----
        const float* Asr = &As[cur][(wave * 16 + lrow) * APAD + khalf];
        const float* Bpr = &Bp[cur][(khalf >> 1) * 256 + 2 * lrow];
        #pragma unroll
        for (int kk = 0; kk < KC; kk += 4) {
            v2f a = *(const v2f*)(Asr + kk);
            #pragma unroll
            for (int n = 0; n < 8; ++n) {
                v2f b = *(const v2f*)(Bpr + (kk >> 1) * 256 + 2 * (n * 16));
                acc[n] = __builtin_amdgcn_wmma_f32_16x16x4_f32(
                    false, a, false, b, (short)0, acc[n], false, false);
            }
        }

        // ---- finish staging next chunk, then swap ----
        if (more) {
            #pragma unroll
            for (int i = 0; i < 4; ++i) {
                int f  = tid + i * 256;
                int k  = f >> 5;
                int c4 = f & 31;
                float* d = &Bp[nxt][(k >> 1) * 256 + (k & 1)];
                d[2 * (c4 * 4 + 0)] = breg[i].x;
                d[2 * (c4 * 4 + 1)] = breg[i].y;
                d[2 * (c4 * 4 + 2)] = breg[i].z;
                d[2 * (c4 * 4 + 3)] = breg[i].w;
            }
#ifdef HAVE_ASYNC_LDS
            async_wait0();
#endif
        }
        __syncthreads();
    }

    // ---- epilogue: element (vgpr v, lane L): M = v + 8*(L>>4), N = L&15 ----
    const int mbase = m0 + wave * 16 + (lane >> 4) * 8;
    #pragma unroll
    for (int n = 0; n < 8; ++n) {
        float bval = use_bias ? bias[n * 16 + lrow] : 0.0f;
        #pragma unroll
        for (int v = 0; v < 8; ++v) {
            int r = mbase + v;
            if (r < M) {
                float x = acc[n][v] + bval;
                if (use_relu) x = fmaxf(x, 0.0f);
                C[(size_t)r * HID + n * 16 + lrow] = x;
            }
        }
    }
}

// ---------------------------------------------------------------------------
// Degree / normalization.
// ---------------------------------------------------------------------------
__global__ void init_deg(float* __restrict__ deg, int n) {
    int i = blockIdx.x * blockDim.x + threadIdx.x;
    if (i < n) deg[i] = 1.0f;              // self-loop contributes 1
}

__global__ void count_deg(const long long* __restrict__ dst,
                          float* __restrict__ deg, int E) {
    int i = blockIdx.x * blockDim.x + threadIdx.x;
    if (i < E) atomicAdd(&deg[(int)dst[i]], 1.0f);
}

__global__ void make_dinv(float* __restrict__ deg, int n) {
    int i = blockIdx.x * blockDim.x + threadIdx.x;
    if (i < n) deg[i] = rsqrtf(deg[i]);    // deg >= 1 always
}

// ---------------------------------------------------------------------------
// Edge scatter: out[dst] += t[src] * dinv[src]*dinv[dst].  One wave per edge,
// float4 per lane; t/out rows (25.6 MB) are resident in the 192 MB L2.
// ---------------------------------------------------------------------------
__global__ __launch_bounds__(256)
void edge_scatter(const long long* __restrict__ src,
                  const long long* __restrict__ dst,
                  const float* __restrict__ dinv,
                  const float* __restrict__ t,
                  float* __restrict__ out, int E)
{
    int wave = blockIdx.x * (blockDim.x >> 5) + (threadIdx.x >> 5);
    int lane = threadIdx.x & 31;
    if (wave >= E) return;
    int s = (int)src[wave];
    int d = (int)dst[wave];
    float* orow = out + (size_t)d * HID + lane * 4;
    __builtin_prefetch(orow, 1, 1);        // global_prefetch_b8 (RMW target)
    float norm = dinv[s] * dinv[d];
    float4 v = ((const float4*)(t + (size_t)s * HID))[lane];
    atomicAdd(orow + 0, v.x * norm);
    atomicAdd(orow + 1, v.y * norm);
    atomicAdd(orow + 2, v.z * norm);
    atomicAdd(orow + 3, v.w * norm);
}

// agg = (maybe relu)(agg + t*dinv^2 + bias)   -- self-loop term + GCN bias
__global__ void finalize_conv(float* __restrict__ agg,
                              const float* __restrict__ t,
                              const float* __restrict__ dinv,
                              const float* __restrict__ bias,
                              int n, int use_relu)
{
    size_t i = (size_t)blockIdx.x * blockDim.x + threadIdx.x;
    if (i >= (size_t)n * HID) return;
    int node = (int)(i >> 7);
    int f    = (int)(i & (HID - 1));
    float di = dinv[node];
    float x  = agg[i] + t[i] * di * di + bias[f];
    if (use_relu) x = fmaxf(x, 0.0f);
    agg[i] = x;
}

// ---------------------------------------------------------------------------
// Global mean pool (batch sorted, values in [0, 128)).
// ---------------------------------------------------------------------------
__global__ __launch_bounds__(256)
void pool_accum(const float* __restrict__ h, const long long* __restrict__ batch,
                float* __restrict__ gsum, float* __restrict__ gcnt, int n)
{
    int wave = blockIdx.x * (blockDim.x >> 5) + (threadIdx.x >> 5);
    int lane = threadIdx.x & 31;
    if (wave >= n) return;
    int g = (int)batch[wave];
    float4 v = ((const float4*)(h + (size_t)wave * HID))[lane];
    float* o = gsum + (size_t)g * HID + lane * 4;
    atomicAdd(o + 0, v.x);
    atomicAdd(o + 1, v.y);
    atomicAdd(o + 2, v.z);
    atomicAdd(o + 3, v.w);
    if (lane == 0) atomicAdd(&gcnt[g], 1.0f);
}

__global__ void pool_div(float* __restrict__ g, const float* __restrict__ gsum,
                         const float* __restrict__ gcnt, int ngraphs)
{
    int i = blockIdx.x * blockDim.x + threadIdx.x;
    if (i >= ngraphs * HID) return;
    g[i] = gsum[i] / fmaxf(gcnt[i >> 7], 1.0f);
}

// out[128 x 3] = g1[128 x 128] @ W_l2[128 x 3] + b_l2
__global__ void head_final(const float* __restrict__ g1,
                           const float* __restrict__ W,
                           const float* __restrict__ b,
                           float* __restrict__ out, int ngraphs)
{
    int t = blockIdx.x * blockDim.x + threadIdx.x;
    if (t >= ngraphs) return;
    float a0 = b[0], a1 = b[1], a2 = b[2];
    for (int k = 0; k < HID; ++k) {
        float gv = g1[(size_t)t * HID + k];
        a0 += gv * W[k * 3 + 0];
        a1 += gv * W[k * 3 + 1];
        a2 += gv * W[k * 3 + 2];
    }
    out[t * 3 + 0] = a0;
    out[t * 3 + 1] = a1;
    out[t * 3 + 2] = a2;
}

// ---------------------------------------------------------------------------
// Host-side orchestration (graph-capture safe: only async stream ops).
// ---------------------------------------------------------------------------
extern "C" void kernel_launch(void* const* d_in, const int* in_sizes, int n_in,
                              void* d_out, int out_size, void* d_ws, size_t ws_size,
                              hipStream_t stream)
{
    const float*     x     = (const float*)d_in[0];
    const long long* eidx  = (const long long*)d_in[1];   // [2, E] int64
    const long long* batch = (const long long*)d_in[2];
    const float* W_emb = (const float*)d_in[3];
    const float* b_emb = (const float*)d_in[4];
    const float* W_c1  = (const float*)d_in[5];
    const float* b_c1  = (const float*)d_in[6];
    const float* W_c2  = (const float*)d_in[7];
    const float* b_c2  = (const float*)d_in[8];
    const float* W_l1  = (const float*)d_in[9];
    const float* b_l1  = (const float*)d_in[10];
    const float* W_l2  = (const float*)d_in[11];
    const float* b_l2  = (const float*)d_in[12];
    float* out = (float*)d_out;

    const int N = in_sizes[2];              // 50000 nodes
    const int E = in_sizes[1] / 2;          // 800000 edges
    const int K_in = in_sizes[3] / HID;     // 768
    const int G = 128;                      // graphs

    const long long* src = eidx;
    const long long* dst = eidx + E;

    // ---- workspace carve-up ----
    char* ws = (char*)d_ws;
    size_t nh = (size_t)N * HID * sizeof(float);          // 25.6 MB each
    float* buf0 = (float*)(ws);                           // h / t2
    float* buf1 = (float*)(ws + nh);                      // t1 / agg2(h2)
    float* buf2 = (float*)(ws + 2 * nh);                  // agg1(h1)
    float* dinv = (float*)(ws + 3 * nh);                  // N floats
    char*  ws2  = ws + 3 * nh + (((size_t)N * sizeof(float) + 255) & ~(size_t)255);
    float* gsum = (float*)(ws2);                          // 128*128
    float* gcnt = (float*)(ws2 + G * HID * sizeof(float));
    float* gbuf = (float*)(ws2 + (G * HID + G) * sizeof(float));
    float* g1   = (float*)(ws2 + (2 * G * HID + G) * sizeof(float));

    const int gemm_blocks = (N + TM - 1) / TM;            // 391
    const int elem_blocks = (int)(((size_t)N * HID + 255) / 256);
    const int edge_waves_blocks = (E + 7) / 8;            // 8 waves / block
    const int node_waves_blocks = (N + 7) / 8;

    // ---- degree / dinv ----
    init_deg<<<(N + 255) / 256, 256, 0, stream>>>(dinv, N);
    count_deg<<<(E + 255) / 256, 256, 0, stream>>>(dst, dinv, E);
    make_dinv<<<(N + 255) / 256, 256, 0, stream>>>(dinv, N);

    // ---- embedding: buf0 = x @ W_emb + b_emb ----
    wmma_gemm_f32<<<gemm_blocks, 256, 0, stream>>>(x, W_emb, b_emb, buf0,
                                                   N, K_in, 1, 0);

    // ---- conv1: t1 = buf0 @ W_c1 -> buf1 ; agg -> buf2 ; relu ----
    wmma_gemm_f32<<<gemm_blocks, 256, 0, stream>>>(buf0, W_c1, b_c1, buf1,
                                                   N, HID, 0, 0);
    (void)hipMemsetAsync(buf2, 0, nh, stream);
    edge_scatter<<<edge_waves_blocks, 256, 0, stream>>>(src, dst, dinv, buf1,
                                                        buf2, E);
    finalize_conv<<<elem_blocks, 256, 0, stream>>>(buf2, buf1, dinv, b_c1, N, 1);

    // ---- conv2: t2 = buf2 @ W_c2 -> buf0 ; agg -> buf1 ; no relu ----
    wmma_gemm_f32<<<gemm_blocks, 256, 0, stream>>>(buf2, W_c2, b_c2, buf0,
                                                   N, HID, 0, 0);
    (void)hipMemsetAsync(buf1, 0, nh, stream);
    edge_scatter<<<edge_waves_blocks, 256, 0, stream>>>(src, dst, dinv, buf0,
                                                        buf1, E);
    finalize_conv<<<elem_blocks, 256, 0, stream>>>(buf1, buf0, dinv, b_c2, N, 0);

    // ---- global mean pool ----
    (void)hipMemsetAsync(gsum, 0, (size_t)G * HID * sizeof(float), stream);
    (void)hipMemsetAsync(gcnt, 0, (size_t)G * sizeof(float), stream);
    pool_accum<<<node_waves_blocks, 256, 0, stream>>>(buf1, batch, gsum, gcnt, N);
    pool_div<<<(G * HID + 255) / 256, 256, 0, stream>>>(gbuf, gsum, gcnt, G);

    // ---- head: g1 = relu(g @ W_l1 + b_l1) (WMMA); out = g1 @ W_l2 + b_l2 ----
    wmma_gemm_f32<<<1, 256, 0, stream>>>(gbuf, W_l1, b_l1, g1, G, HID, 1, 1);
    head_final<<<1, 128, 0, stream>>>(g1, W_l2, b_l2, out, G);
}